// BatchDistance_8555574853751
// MI455X (gfx1250) — compile-verified
//
#include <hip/hip_runtime.h>

typedef __attribute__((ext_vector_type(2))) float v2f;
typedef __attribute__((ext_vector_type(8))) float v8f;

#define D_FEAT     64
#define LDS_STRIDE 68   // 64 + 4 pad: conflict-free WMMA frag loads, 16B-aligned float4 stores

// Output D = -(L*A^T + A*L^T), L = x*log(x); full symmetric 2048x2048 fp32.
// Workgroup: 256 threads (8 wave32 waves) -> 32x64 output tile, 16x16 WMMA tile per wave.
__global__ __launch_bounds__(256)
void batchdist_wmma_f32(const float* __restrict__ X, float* __restrict__ out, int n)
{
    __shared__ float sAi[32 * LDS_STRIDE];
    __shared__ float sLi[32 * LDS_STRIDE];
    __shared__ float sAj[64 * LDS_STRIDE];
    __shared__ float sLj[64 * LDS_STRIDE];

    const int tid = threadIdx.x;
    const int i0  = blockIdx.y * 32;   // row block
    const int j0  = blockIdx.x * 64;   // col block

    // ---- cooperative, coalesced global loads; compute L = a*log(a) on the fly ----
    // i-block: 32 rows x 16 float4 = 512 quads (2 per thread)
    for (int q = tid; q < 32 * 16; q += 256) {
        const int row = q >> 4;
        const int c4  = (q & 15) << 2;
        const float4 g = *(const float4*)(X + (size_t)(i0 + row) * D_FEAT + c4);
        float4 l;
        l.x = g.x * __logf(g.x); l.y = g.y * __logf(g.y);
        l.z = g.z * __logf(g.z); l.w = g.w * __logf(g.w);
        *(float4*)(sAi + row * LDS_STRIDE + c4) = g;
        *(float4*)(sLi + row * LDS_STRIDE + c4) = l;
    }
    // j-block: 64 rows x 16 float4 = 1024 quads (4 per thread)
    for (int q = tid; q < 64 * 16; q += 256) {
        const int row = q >> 4;
        const int c4  = (q & 15) << 2;
        const float4 g = *(const float4*)(X + (size_t)(j0 + row) * D_FEAT + c4);
        float4 l;
        l.x = g.x * __logf(g.x); l.y = g.y * __logf(g.y);
        l.z = g.z * __logf(g.z); l.w = g.w * __logf(g.w);
        *(float4*)(sAj + row * LDS_STRIDE + c4) = g;
        *(float4*)(sLj + row * LDS_STRIDE + c4) = l;
    }
    __syncthreads();

    // ---- per-wave 16x16 WMMA tile ----
    const int wave = tid >> 5;
    const int lane = tid & 31;
    const int half = lane >> 4;     // selects K pair within a 16x4 fragment
    const int mn   = lane & 15;     // A-row (for A frags) / B-col (for B frags)
    const int sr   = wave >> 2;     // 0..1  sub-tile row
    const int sc   = wave & 3;      // 0..3  sub-tile col

    // ISA 7.12.2 fp32 A 16x4: lane(h,m) VGPR v holds element (m, k0 + 2h + v);
    // B 4x16 mirrors with n = lane&15.
    const float* li = sLi + (sr * 16 + mn) * LDS_STRIDE + 2 * half;
    const float* ai = sAi + (sr * 16 + mn) * LDS_STRIDE + 2 * half;
    const float* aj = sAj + (sc * 16 + mn) * LDS_STRIDE + 2 * half;
    const float* lj = sLj + (sc * 16 + mn) * LDS_STRIDE + 2 * half;

    v8f c = {};
#pragma unroll
    for (int k = 0; k < D_FEAT; k += 4) {
        const v2f aL = *(const v2f*)(li + k);   // L_i fragment
        const v2f aA = *(const v2f*)(ai + k);   // A_i fragment
        const v2f bA = *(const v2f*)(aj + k);   // A_j fragment (acts as A^T)
        const v2f bL = *(const v2f*)(lj + k);   // L_j fragment (acts as L^T)
        // c += L_i * A_j^T   then   c += A_i * L_j^T
        c = __builtin_amdgcn_wmma_f32_16x16x4_f32(false, aL, false, bA,
                                                  (short)0, c, false, false);
        c = __builtin_amdgcn_wmma_f32_16x16x4_f32(false, aA, false, bL,
                                                  (short)0, c, false, false);
    }

    // ---- store D = -c ; C/D layout: VGPR r -> row (r + 8*half), col = mn ----
    float* orow = out + (size_t)(i0 + sr * 16 + 8 * half) * n + (j0 + sc * 16 + mn);
#pragma unroll
    for (int r = 0; r < 8; ++r) {
        orow[(size_t)r * n] = -c[r];
    }
}

extern "C" void kernel_launch(void* const* d_in, const int* in_sizes, int n_in,
                              void* d_out, int out_size, void* d_ws, size_t ws_size,
                              hipStream_t stream) {
    (void)n_in; (void)out_size; (void)d_ws; (void)ws_size;
    const float* x1 = (const float*)d_in[0];
    float* out = (float*)d_out;
    const int n = in_sizes[0] / D_FEAT;          // 2048
    dim3 grid(n / 64, n / 32);                   // (32, 64)
    batchdist_wmma_f32<<<grid, 256, 0, stream>>>(x1, out, n);
}